// PureFEPLayer_4750233829621
// MI455X (gfx1250) — compile-verified
//
#include <hip/hip_runtime.h>
#include <math.h>

// ---------------- problem constants ----------------
#define NAG   512          // N agents
#define KD    127          // K
#define KP    128          // padded K
#define BIG   (NAG*KP*KP)  // one batched 128x128 buffer, in floats

#define ALPHA   0.01f
#define LAMBDA  1.0f
#define KAPPA   0.1f
#define MU_LR   0.1f
#define SIG_LR  0.025f
#define EPSV    1e-6f

typedef __attribute__((ext_vector_type(2))) float v2f;
typedef __attribute__((ext_vector_type(8))) float v8f;
typedef __attribute__((ext_vector_type(4))) unsigned int u32x4;
typedef __attribute__((ext_vector_type(8))) int i32x8;
typedef __attribute__((ext_vector_type(4))) int i32x4;

// ---------------------------------------------------------------------------
// Tensor Data Mover: 2D tile load global -> LDS (descriptor per ISA 08 §8)
//   tile_d1 rows of tile_d0 elements (fp32), row stride = stride0 elements.
//   LDS receives the tile packed contiguously (row-major tile_d1 x tile_d0).
//   6-arg builtin form (clang-23 / therock headers).
// ---------------------------------------------------------------------------
static __device__ __forceinline__ void tdm_load_2d(unsigned lds_addr,
                                                   const float* gptr,
                                                   unsigned tile_d0, unsigned tile_d1,
                                                   unsigned tensor_d0, unsigned tensor_d1,
                                                   unsigned stride0) {
  unsigned long long ga = (unsigned long long)(uintptr_t)gptr;
  u32x4 g0;
  g0.x = 1u;                                            // count=1, user D#
  g0.y = lds_addr;                                      // lds_addr [63:32]
  g0.z = (unsigned)(ga & 0xffffffffu);                  // global_addr lo
  g0.w = (unsigned)((ga >> 32) & 0x01ffffffu) | 0x80000000u; // addr hi | type=2
  i32x8 g1;
  g1[0] = (int)(2u << 16);                              // data_size = 4 bytes
  g1[1] = (int)((tensor_d0 & 0xffffu) << 16);           // tensor_dim0 [63:48]
  g1[2] = (int)((tensor_d0 >> 16) | ((tensor_d1 & 0xffffu) << 16)); // [79:64],[95:80]
  g1[3] = (int)((tensor_d1 >> 16) | (tile_d0 << 16));   // [111:96], tile_dim0 [127:112]
  g1[4] = (int)(tile_d1 & 0xffffu);                     // tile_dim1 [143:128]
  g1[5] = (int)stride0;                                 // tensor_dim0_stride lo32
  g1[6] = 0;
  g1[7] = 0;
  i32x4 z4 = {0, 0, 0, 0};
  i32x8 z8 = {0, 0, 0, 0, 0, 0, 0, 0};
  __builtin_amdgcn_tensor_load_to_lds(g0, g1, z4, z4, z8, 0);
}

// ---------------------------------------------------------------------------
// fp32 WMMA 16x16 tile helpers (V_WMMA_F32_16X16X4_F32, wave32)
//   A 16x4 fragment : lane L -> M=L%16, VGPR0=K(2*kh), VGPR1=K(2*kh+1), kh=L/16
//   B 4x16 fragment : lane L -> N=L%16, VGPR0=K(2*kh), VGPR1=K(2*kh+1)
//   C/D             : 8 VGPRs, row M = r + 8*kh, col N = L%16
// ---------------------------------------------------------------------------
// acc = A[tile 16 x len] * B[len x tile 16], B row-major B[k*ldb + n]
static __device__ __forceinline__ v8f wmma_tile(const float* __restrict__ A, int lda,
                                                const float* __restrict__ B, int ldb,
                                                int len, int lane) {
  const int n16 = lane & 15;
  const int kh  = lane >> 4;
  v8f acc = {};
#pragma unroll 4
  for (int kk = 0; kk < len; kk += 4) {
    const int k0 = kk + 2 * kh;
    v2f a, b;
    a.x = A[n16 * lda + k0];
    a.y = A[n16 * lda + k0 + 1];
    b.x = B[k0 * ldb + n16];
    b.y = B[(k0 + 1) * ldb + n16];
    acc = __builtin_amdgcn_wmma_f32_16x16x4_f32(false, a, false, b,
                                                (short)0, acc, false, false);
  }
  return acc;
}

// acc = A[16 x len] * Bt[16 x len]^T   (B transposed: element (k,n) = Bt[n*ldb + k])
static __device__ __forceinline__ v8f wmma_tile_bt(const float* __restrict__ A, int lda,
                                                   const float* __restrict__ Bt, int ldb,
                                                   int len, int lane) {
  const int n16 = lane & 15;
  const int kh  = lane >> 4;
  v8f acc = {};
#pragma unroll 4
  for (int kk = 0; kk < len; kk += 4) {
    const int k0 = kk + 2 * kh;
    v2f a, b;
    a.x = A[n16 * lda + k0];
    a.y = A[n16 * lda + k0 + 1];
    b.x = Bt[n16 * ldb + k0];
    b.y = Bt[n16 * ldb + k0 + 1];
    acc = __builtin_amdgcn_wmma_f32_16x16x4_f32(false, a, false, b,
                                                (short)0, acc, false, false);
  }
  return acc;
}

// ---------------------------------------------------------------------------
// 1) A_n = (phi.G)/2^4 padded to 128x128 ; E_n = I
// ---------------------------------------------------------------------------
__global__ void k_build_A(const float* __restrict__ phi, const float* __restrict__ G,
                          float* __restrict__ Abuf, float* __restrict__ Ebuf) {
  const int n = blockIdx.x;
  const float p0 = phi[n * 3 + 0], p1 = phi[n * 3 + 1], p2 = phi[n * 3 + 2];
  const float s = 1.0f / 16.0f;   // fold 2^-squarings scaling
  float* An = Abuf + (size_t)n * KP * KP;
  float* En = Ebuf + (size_t)n * KP * KP;
  for (int idx = threadIdx.x; idx < KP * KP; idx += blockDim.x) {
    const int k = idx >> 7, l = idx & 127;
    float v = 0.0f;
    if (k < KD && l < KD) {
      const int o = k * KD + l;
      v = (p0 * G[o] + p1 * G[KD * KD + o] + p2 * G[2 * KD * KD + o]) * s;
    }
    An[idx] = v;
    En[idx] = (k == l) ? 1.0f : 0.0f;
  }
}

// ---------------------------------------------------------------------------
// 2) per-agent stats: padded mu / clipped sigma / 1/sigma ; logdet
// ---------------------------------------------------------------------------
__global__ void k_stats(const float* __restrict__ mu_q, const float* __restrict__ sigma_q,
                        float* __restrict__ mu_pad, float* __restrict__ sq_pad,
                        float* __restrict__ inv_pad, float* __restrict__ logdet) {
  const int n = blockIdx.x;
  const int t = threadIdx.x;          // 0..127
  float mu = 0.f, sq = 0.f, inv = 0.f, lg = 0.f;
  if (t < KD) {
    mu = mu_q[n * KD + t];
    sq = fmaxf(sigma_q[n * KD + t], EPSV);
    inv = 1.0f / sq;
    lg = __logf(sq);
  }
  mu_pad[n * KP + t] = mu;
  sq_pad[n * KP + t] = sq;
  inv_pad[n * KP + t] = inv;
  __shared__ float red[KP];
  red[t] = lg;
  __syncthreads();
  for (int s = 64; s > 0; s >>= 1) {
    if (t < s) red[t] += red[t + s];
    __syncthreads();
  }
  if (t == 0) logdet[n] = red[0];
}

// ---------------------------------------------------------------------------
// 3) batched 128x128x128 GEMM: O_n = addI*I + scale*(A_n @ B_n)
//    TDM double-buffered LDS staging (16-wide K chunks) + WMMA f32 from LDS.
//    One block per agent; wave w owns output row-panel w (16 x 128 in regs).
// ---------------------------------------------------------------------------
#define KCH 16
__global__ void k_expm_step(const float* __restrict__ Abuf, const float* __restrict__ Bbuf,
                            float* __restrict__ Obuf, float scale, int addI) {
  const int n = blockIdx.x;
  const int wave = threadIdx.x >> 5;
  const int lane = threadIdx.x & 31;
  const int n16 = lane & 15, kh = lane >> 4;
  __shared__ float As[2][KP * KCH];   // A[:, kc:kc+16]  -> As[m][kk]
  __shared__ float Bs[2][KCH * KP];   // B[kc:kc+16, :]  -> Bs[kk][col]
  const float* A = Abuf + (size_t)n * KP * KP;
  const float* B = Bbuf + (size_t)n * KP * KP;
  float* O = Obuf + (size_t)n * KP * KP;

  v8f acc[8] = {};

  // prologue: stage chunk 0
  if (wave == 0) {
    tdm_load_2d((unsigned)(uintptr_t)&As[0][0], A, KCH, KP, KP, KP, KP);
    tdm_load_2d((unsigned)(uintptr_t)&Bs[0][0], B, KP, KCH, KP, KP, KP);
    __builtin_amdgcn_s_wait_tensorcnt(0);
  }
  __syncthreads();

  int cur = 0;
  for (int kc = 0; kc < KP; kc += KCH) {
    // async-prefetch next chunk into the other buffer while we compute
    if (wave == 0 && (kc + KCH) < KP) {
      tdm_load_2d((unsigned)(uintptr_t)&As[cur ^ 1][0], A + (kc + KCH), KCH, KP, KP, KP, KP);
      tdm_load_2d((unsigned)(uintptr_t)&Bs[cur ^ 1][0], B + (size_t)(kc + KCH) * KP,
                  KP, KCH, KP, KP, KP);
    }
    const float* as = &As[cur][0];
    const float* bs = &Bs[cur][0];
#pragma unroll
    for (int tn = 0; tn < 8; ++tn) {
#pragma unroll
      for (int kk = 0; kk < KCH; kk += 4) {
        const int k0 = kk + 2 * kh;
        v2f a, b;
        a.x = as[(wave * 16 + n16) * KCH + k0];
        a.y = as[(wave * 16 + n16) * KCH + k0 + 1];
        b.x = bs[k0 * KP + tn * 16 + n16];
        b.y = bs[(k0 + 1) * KP + tn * 16 + n16];
        acc[tn] = __builtin_amdgcn_wmma_f32_16x16x4_f32(false, a, false, b,
                                                        (short)0, acc[tn], false, false);
      }
    }
    if (wave == 0) __builtin_amdgcn_s_wait_tensorcnt(0);
    __syncthreads();
    cur ^= 1;
  }

#pragma unroll
  for (int tn = 0; tn < 8; ++tn) {
#pragma unroll
    for (int r = 0; r < 8; ++r) {
      const int row = wave * 16 + r + 8 * kh;
      const int col = tn * 16 + n16;
      float v = acc[tn][r] * scale;
      if (addI && row == col) v += 1.0f;
      O[row * KP + col] = v;
    }
  }
}

// ---------------------------------------------------------------------------
// 4) a_n = R_n^T mu_n  (padded)
// ---------------------------------------------------------------------------
__global__ void k_avec(const float* __restrict__ R, const float* __restrict__ mu_pad,
                       float* __restrict__ avec) {
  const int n = blockIdx.x, l = threadIdx.x;
  const float* Rn = R + (size_t)n * KP * KP;
  const float* mn = mu_pad + n * KP;
  float s = 0.f;
  for (int k = 0; k < KP; ++k) s += Rn[k * KP + l] * mn[k];
  avec[n * KP + l] = s;
}

// ---------------------------------------------------------------------------
// 5) klbase[i,j] = 0.5*(trace(i,j) - K + logdet[j] - logdet[i])   (WMMA GEMM)
// ---------------------------------------------------------------------------
__global__ void k_klbase(const float* __restrict__ sq_pad, const float* __restrict__ inv_pad,
                         const float* __restrict__ logdet, float* __restrict__ klbase) {
  const int tile = blockIdx.x * 8 + (threadIdx.x >> 5);  // 0..1023
  const int lane = threadIdx.x & 31;
  const int n16 = lane & 15, kh = lane >> 4;
  const int ti = tile >> 5, tj = tile & 31;
  v8f acc = wmma_tile_bt(sq_pad + ti * 16 * KP, KP, inv_pad + tj * 16 * KP, KP, KP, lane);
#pragma unroll
  for (int r = 0; r < 8; ++r) {
    const int i = ti * 16 + r + 8 * kh;
    const int j = tj * 16 + n16;
    klbase[i * NAG + j] = 0.5f * (acc[r] - (float)KD + logdet[j] - logdet[i]);
  }
}

// ---------------------------------------------------------------------------
// 6) per-agent i: m tiles via WMMA, quad -> kl row -> softmax -> beta row
// ---------------------------------------------------------------------------
__global__ void k_beta(const float* __restrict__ R, const float* __restrict__ avec,
                       const float* __restrict__ mu_pad, const float* __restrict__ inv_pad,
                       const float* __restrict__ klbase, float* __restrict__ beta) {
  const int i = blockIdx.x;
  const int wave = threadIdx.x >> 5, lane = threadIdx.x & 31;
  const int n16 = lane & 15, kh = lane >> 4;
  __shared__ float part[8][16];
  __shared__ float klrow[NAG];
  __shared__ float red2[256];
  const float* Ri = R + (size_t)i * KP * KP;
  const float* mui = mu_pad + i * KP;

  for (int jt = 0; jt < 32; ++jt) {
    // m[k, j] = sum_l Ri[k,l] * avec[j,l] for k in wave's 16 rows, 16 j's
    v8f acc = wmma_tile_bt(Ri + wave * 16 * KP, KP, avec + (jt * 16) * KP, KP, KP, lane);
    const int j = jt * 16 + n16;
    float p = 0.f;
#pragma unroll
    for (int r = 0; r < 8; ++r) {
      const int k = wave * 16 + r + 8 * kh;
      const float d = mui[k] - acc[r];
      p += d * d * inv_pad[j * KP + k];
    }
    p += __shfl_xor(p, 16);                 // merge the two k-halves (same j)
    if (lane < 16) part[wave][lane] = p;    // sum over this wave's 16 k rows
    __syncthreads();
    if (threadIdx.x < 16) {
      float q = 0.f;
      for (int w = 0; w < 8; ++w) q += part[w][threadIdx.x];
      const int jj = jt * 16 + threadIdx.x;
      klrow[jj] = klbase[i * NAG + jj] + 0.5f * q;
    }
    __syncthreads();
  }

  // softmax over j of -KAPPA*kl
  float mx = -1e30f;
  for (int j = threadIdx.x; j < NAG; j += 256) mx = fmaxf(mx, -KAPPA * klrow[j]);
  red2[threadIdx.x] = mx;
  __syncthreads();
  for (int s = 128; s > 0; s >>= 1) {
    if (threadIdx.x < s) red2[threadIdx.x] = fmaxf(red2[threadIdx.x], red2[threadIdx.x + s]);
    __syncthreads();
  }
  mx = red2[0];
  __syncthreads();
  float sm = 0.f;
  for (int j = threadIdx.x; j < NAG; j += 256) {
    const float e = __expf(-KAPPA * klrow[j] - mx);
    klrow[j] = e;
    sm += e;
  }
  red2[threadIdx.x] = sm;
  __syncthreads();
  for (int s = 128; s > 0; s >>= 1) {
    if (threadIdx.x < s) red2[threadIdx.x] += red2[threadIdx.x + s];
    __syncthreads();
  }
  const float inv = 1.0f / red2[0];
  for (int j = threadIdx.x; j < NAG; j += 256) beta[i * NAG + j] = klrow[j] * inv;
}

// ---------------------------------------------------------------------------
// 7) ssig[i,k] = sum_j beta[i,j] * inv_pad[j,k]   (WMMA GEMM, contraction 512)
// ---------------------------------------------------------------------------
__global__ void k_ssig(const float* __restrict__ beta, const float* __restrict__ inv_pad,
                       float* __restrict__ ssig) {
  const int tile = blockIdx.x * 8 + (threadIdx.x >> 5);  // 0..255
  const int lane = threadIdx.x & 31;
  const int n16 = lane & 15, kh = lane >> 4;
  const int ti = tile >> 3, tk = tile & 7;
  v8f acc = wmma_tile(beta + ti * 16 * NAG, NAG, inv_pad + tk * 16, KP, NAG, lane);
#pragma unroll
  for (int r = 0; r < 8; ++r) {
    const int i = ti * 16 + r + 8 * kh;
    const int k = tk * 16 + n16;
    ssig[i * KP + k] = acc[r];
  }
}

// ---------------------------------------------------------------------------
// 8) tmu[i,k] = sum_j beta[i,j]*inv_s[j,k]*m[i,j,k]  (recompute m tiles, WMMA)
// ---------------------------------------------------------------------------
__global__ void k_tmu(const float* __restrict__ R, const float* __restrict__ avec,
                      const float* __restrict__ inv_pad, const float* __restrict__ beta,
                      float* __restrict__ tmu) {
  const int i = blockIdx.x;
  const int wave = threadIdx.x >> 5, lane = threadIdx.x & 31;
  const int n16 = lane & 15, kh = lane >> 4;
  __shared__ float tacc[KP];
  for (int t = threadIdx.x; t < KP; t += 256) tacc[t] = 0.f;
  __syncthreads();
  const float* Ri = R + (size_t)i * KP * KP;
  for (int jt = 0; jt < 32; ++jt) {
    v8f acc = wmma_tile_bt(Ri + wave * 16 * KP, KP, avec + (jt * 16) * KP, KP, KP, lane);
    const int j = jt * 16 + n16;
    const float bw = beta[i * NAG + j];
#pragma unroll
    for (int r = 0; r < 8; ++r) {
      const int k = wave * 16 + r + 8 * kh;
      float c = bw * inv_pad[j * KP + k] * acc[r];
      // sum over the 16 j's of each half (halves keep distinct k)
      c += __shfl_xor(c, 1);
      c += __shfl_xor(c, 2);
      c += __shfl_xor(c, 4);
      c += __shfl_xor(c, 8);
      if (n16 == 0) tacc[k] += c;   // lanes 0 and 16 -> disjoint k, no race
    }
  }
  __syncthreads();
  for (int t = threadIdx.x; t < KP; t += 256) tmu[i * KP + t] = tacc[t];
}

// ---------------------------------------------------------------------------
// 9) zero the two global-norm accumulators
// ---------------------------------------------------------------------------
__global__ void k_zero(float* __restrict__ norms) {
  if (threadIdx.x < 2) norms[threadIdx.x] = 0.f;
}

// ---------------------------------------------------------------------------
// 10) natural gradients + global sum-of-squares
// ---------------------------------------------------------------------------
__global__ void k_grads(const float* __restrict__ mu_pad, const float* __restrict__ sq_pad,
                        const float* __restrict__ inv_pad, const float* __restrict__ mu_p,
                        const float* __restrict__ sigma_p, const float* __restrict__ ssig,
                        const float* __restrict__ tmu, float* __restrict__ ng_mu,
                        float* __restrict__ ng_sig, float* __restrict__ norms) {
  const int idx = blockIdx.x * blockDim.x + threadIdx.x;  // over N*KP
  const int n = idx >> 7, k = idx & 127;
  float gm = 0.f, gs = 0.f;
  if (k < KD) {
    const float sp = fmaxf(sigma_p[n * KD + k], EPSV);
    const float mu = mu_pad[idx], sq = sq_pad[idx], inv = inv_pad[idx];
    const float smu = mu * ssig[idx] - tmu[idx];       // sum_j beta*diff*inv_s
    const float grad_mu = ALPHA * (mu - mu_p[n * KD + k]) / sp + LAMBDA * smu;
    const float grad_sg = ALPHA * 0.5f * (1.0f / sp - inv) + LAMBDA * 0.5f * (ssig[idx] - inv);
    gm = grad_mu * sq;
    gs = grad_sg * 2.0f * sq * sq;
  }
  ng_mu[idx] = gm;
  ng_sig[idx] = gs;
  __shared__ float r0[256], r1[256];
  r0[threadIdx.x] = gm * gm;
  r1[threadIdx.x] = gs * gs;
  __syncthreads();
  for (int s = 128; s > 0; s >>= 1) {
    if (threadIdx.x < s) {
      r0[threadIdx.x] += r0[threadIdx.x + s];
      r1[threadIdx.x] += r1[threadIdx.x + s];
    }
    __syncthreads();
  }
  if (threadIdx.x == 0) {
    atomicAdd(&norms[0], r0[0]);
    atomicAdd(&norms[1], r1[0]);
  }
}

// ---------------------------------------------------------------------------
// 11) clipped update -> d_out = [mu_new (N*K) | sigma_new (N*K)]
// ---------------------------------------------------------------------------
__global__ void k_update(const float* __restrict__ mu_pad, const float* __restrict__ sq_pad,
                         const float* __restrict__ ng_mu, const float* __restrict__ ng_sig,
                         const float* __restrict__ norms, float* __restrict__ out) {
  const int idx = blockIdx.x * blockDim.x + threadIdx.x;
  if (idx >= NAG * KD) return;
  const int n = idx / KD, k = idx % KD;
  const int p = n * KP + k;
  const float cm = fminf(1.0f, 1.0f / fmaxf(sqrtf(norms[0]), EPSV));
  const float cs = fminf(1.0f, 1.0f / fmaxf(sqrtf(norms[1]), EPSV));
  out[idx] = mu_pad[p] - MU_LR * ng_mu[p] * cm;
  out[NAG * KD + idx] = fmaxf(sq_pad[p] - SIG_LR * ng_sig[p] * cs, EPSV);
}

// ---------------------------------------------------------------------------
extern "C" void kernel_launch(void* const* d_in, const int* in_sizes, int n_in,
                              void* d_out, int out_size, void* d_ws, size_t ws_size,
                              hipStream_t stream) {
  const float* mu_q    = (const float*)d_in[0];
  const float* sigma_q = (const float*)d_in[1];
  const float* mu_p    = (const float*)d_in[2];
  const float* sigma_p = (const float*)d_in[3];
  const float* phi     = (const float*)d_in[4];
  const float* gens    = (const float*)d_in[5];
  float* out = (float*)d_out;

  float* ws = (float*)d_ws;
  float* BUF0 = ws;                   // A (later reused for klbase/beta)
  float* BUF1 = ws + (size_t)BIG;     // E ping
  float* BUF2 = ws + (size_t)2 * BIG; // E pong
  float* mu_pad  = ws + (size_t)3 * BIG;
  float* sq_pad  = mu_pad + NAG * KP;
  float* inv_pad = sq_pad + NAG * KP;
  float* avec    = inv_pad + NAG * KP;
  float* ssig    = avec + NAG * KP;
  float* tmu     = ssig + NAG * KP;
  float* ng_mu   = tmu + NAG * KP;
  float* ng_sig  = ng_mu + NAG * KP;
  float* logdet  = ng_sig + NAG * KP;
  float* norms   = logdet + NAG;

  // A_n (scaled), E = I, per-agent stats
  k_build_A<<<NAG, 256, 0, stream>>>(phi, gens, BUF0, BUF1);
  k_stats<<<NAG, 128, 0, stream>>>(mu_q, sigma_q, mu_pad, sq_pad, inv_pad, logdet);

  // expm: Horner (12) + squarings (4), ping-pong E<->T  (16 swaps -> R = BUF1)
  float* E = BUF1;
  float* T = BUF2;
  for (int k = 12; k >= 1; --k) {
    k_expm_step<<<NAG, 256, 0, stream>>>(BUF0, E, T, 1.0f / (float)k, 1);
    float* t = E; E = T; T = t;
  }
  for (int s = 0; s < 4; ++s) {
    k_expm_step<<<NAG, 256, 0, stream>>>(E, E, T, 1.0f, 0);
    float* t = E; E = T; T = t;
  }
  const float* R = E;

  // a_j = R_j^T mu_j
  k_avec<<<NAG, KP, 0, stream>>>(R, mu_pad, avec);

  // A-buffer is free now: reuse for klbase & beta
  float* klbase = BUF0;
  float* beta   = BUF0 + NAG * NAG;
  k_klbase<<<128, 256, 0, stream>>>(sq_pad, inv_pad, logdet, klbase);
  k_beta<<<NAG, 256, 0, stream>>>(R, avec, mu_pad, inv_pad, klbase, beta);

  // gradient contractions
  k_ssig<<<32, 256, 0, stream>>>(beta, inv_pad, ssig);
  k_tmu<<<NAG, 256, 0, stream>>>(R, avec, inv_pad, beta, tmu);

  // natural gradients, global norm clip, update
  k_zero<<<1, 32, 0, stream>>>(norms);
  k_grads<<<(NAG * KP) / 256, 256, 0, stream>>>(mu_pad, sq_pad, inv_pad, mu_p, sigma_p,
                                                ssig, tmu, ng_mu, ng_sig, norms);
  k_update<<<(NAG * KD + 255) / 256, 256, 0, stream>>>(mu_pad, sq_pad, ng_mu, ng_sig,
                                                       norms, out);
}